// EnhancedFeatureLayer_31061203485277
// MI455X (gfx1250) — compile-verified
//
#include <hip/hip_runtime.h>
#include <stdint.h>
#include <math.h>

#define FEPS 1e-8f
#define NT 128          // threads per block == rows per block
#define RS 135          // LDS row stride in dwords: 120 data + 15 TDM pad (1 per 8)

typedef unsigned int v4u __attribute__((ext_vector_type(4)));
typedef int          v8i __attribute__((ext_vector_type(8)));
typedef int          v4i __attribute__((ext_vector_type(4)));

__device__ __forceinline__ float rcpf(float x) {
#if __has_builtin(__builtin_amdgcn_rcpf)
  return __builtin_amdgcn_rcpf(x);      // v_rcp_f32
#else
  return 1.0f / x;
#endif
}
__device__ __forceinline__ float rsqf(float x) {
#if __has_builtin(__builtin_amdgcn_rsqf)
  return __builtin_amdgcn_rsqf(x);      // v_rsq_f32
#else
  return rsqrtf(x);
#endif
}

#define ACC(i, vv) { const float _v = (vv); sum[i] += _v; sq[i] = fmaf(_v, _v, sq[i]); }

// One streaming pass over s=0..23 for this thread's row.
// P2=false: accumulate per-channel sum/sumsq.  P2=true: recompute + normalize + store.
// Pass-2 store: out = fmaf(f, rstd[ch], nms[ch]) with nms = -mean*rstd.
template <bool P2>
__device__ __forceinline__ void run_pass(const float* __restrict__ rin,
                                         const float* __restrict__ tab,
                                         float2* __restrict__ histp,
                                         float (&sum)[25], float (&sq)[25],
                                         const float (&nms)[25], const float (&rstd)[25],
                                         float* __restrict__ orow) {
  float lc1 = 0.f, lc2 = 0.f, lc3 = 0.f, lc4 = 0.f, lc5 = 0.f, lc6 = 0.f;
  float e3 = 0.f, e6 = 0.f, e12 = 0.f, e24 = 0.f, w6 = 0.f, w12 = 0.f;
  float csg = 0.f, csl = 0.f, g0 = 0.f, l0 = 0.f, obv = 0.f, cprev = 0.f;
#pragma unroll 8
  for (int s = 0; s < 24; ++s) {
    const float* t = &tab[s * 12];
    const float p3 = t[0], p6 = t[1], p12 = t[2], p24 = t[3];
    const float r3 = t[4], r6 = t[5], r12 = t[6], r24 = t[7];
    const int j = s * 5;
    const float o  = rin[j     + ( j      >> 3)];
    const float h  = rin[j + 1 + ((j + 1) >> 3)];
    const float l  = rin[j + 2 + ((j + 2) >> 3)];
    const float c  = rin[j + 3 + ((j + 3) >> 3)];
    const float v  = rin[j + 4 + ((j + 4) >> 3)];

    const float lc  = __logf(c + FEPS);
    const float ret = (s > 0) ? (lc - lc1) : 0.0f;
    const float rcc = rcpf(c + FEPS);
    const float hl  = (h - l) * rcc;
    const float oc  = fabsf(o - c) * rcc;

    // EMA(span): acc = cumsum(x*q^s); ema = acc * (a/(1-q^(s+1)))   [table: q^s, a/(1-q^(s+1))]
    e3  = fmaf(c, p3,  e3 );  const float m3v  = e3  * r3;
    e6  = fmaf(c, p6,  e6 );  const float m6v  = e6  * r6;
    e12 = fmaf(c, p12, e12);  const float m12v = e12 * r12;
    e24 = fmaf(c, p24, e24);  const float m24v = e24 * r24;
    w6  = fmaf(v, p6,  w6 );  const float v6v  = w6  * r6;
    w12 = fmaf(v, p12, w12);  const float v12v = w12 * r12;
    const float rc = rcpf(c);
    const float rv = rcpf(v + FEPS);
    const float f15 = __logf(1.0f + v);            // log1p, v in (0.1, 1.1]
    const float mo3 = (s >= 3) ? (lc - lc3) : 0.0f;
    const float mo6 = (s >= 6) ? (lc - lc6) : 0.0f;
    const float f20 = (c - m12v) * rcpf(m12v + FEPS);

    // RSI streaming: lane-local cumsums; window-14 with left replicate pad.
    const float dlt = (s > 0) ? (c - cprev) : 0.0f;
    csg += fmaxf(dlt, 0.0f);
    csl += fmaxf(-dlt, 0.0f);
    if (!P2) { if (s > 0) histp[(s - 1) * NT] = make_float2(csg, csl); }
    g0 = (s == 1) ? csg : g0;
    l0 = (s == 1) ? csl : l0;
    const int i = s - 1;
    const float2 h14 = histp[((s >= 15) ? (s - 15) : 0) * NT];  // cs[i-14] (pass1-written)
    const float agb = (i <= 13) ? (csg + (float)(13 - i) * g0) : (csg - h14.x);
    const float alb = (i <= 13) ? (csl + (float)(13 - i) * l0) : (csl - h14.y);
    const float ag14 = agb * (1.0f / 14.0f);
    const float al14 = alb * (1.0f / 14.0f);
    // 100 - 100/(1+rs), rs = ag/(al+eps)  ==  100*ag/(al+eps+ag)
    const float rsi = 100.0f * ag14 * rcpf(al14 + FEPS + ag14);  // rsi_raw[s-1], valid s>=1

    const float sg = (dlt > 0.0f) ? 1.0f : ((dlt < 0.0f) ? -1.0f : 0.0f);
    obv = fmaf(v, sg, obv);

    if (!P2) {
      ACC(0, o)  ACC(1, h)  ACC(2, l)  ACC(3, c)  ACC(4, v)
      ACC(5, ret) ACC(6, fabsf(ret)) ACC(7, hl) ACC(8, oc)
      ACC(9,  m3v * rc)  ACC(10, m6v * rc)  ACC(11, m12v * rc)  ACC(12, m24v * rc)
      ACC(13, v6v * rv)  ACC(14, v12v * rv)
      ACC(15, f15) ACC(18, mo3) ACC(19, mo6) ACC(20, f20) ACC(22, obv)
      // ch21 (rsi, edge-padded left by 1): rsi_raw[0] counts twice (slots s=0 and s=1).
      if (s >= 1) {
        const float w = (s == 1) ? 2.0f : 1.0f;
        sum[21] += w * rsi;
        sq[21]  += w * rsi * rsi;
      }
    } else {
      float* op = orow + s * 25;
      op[0]  = fmaf(o,   rstd[0],  nms[0]);
      op[1]  = fmaf(h,   rstd[1],  nms[1]);
      op[2]  = fmaf(l,   rstd[2],  nms[2]);
      op[3]  = fmaf(c,   rstd[3],  nms[3]);
      op[4]  = fmaf(v,   rstd[4],  nms[4]);
      op[5]  = fmaf(ret, rstd[5],  nms[5]);
      op[6]  = fmaf(fabsf(ret), rstd[6], nms[6]);
      op[7]  = fmaf(hl,  rstd[7],  nms[7]);
      op[8]  = fmaf(oc,  rstd[8],  nms[8]);
      op[9]  = fmaf(m3v  * rc, rstd[9],  nms[9]);
      op[10] = fmaf(m6v  * rc, rstd[10], nms[10]);
      op[11] = fmaf(m12v * rc, rstd[11], nms[11]);
      op[12] = fmaf(m24v * rc, rstd[12], nms[12]);
      op[13] = fmaf(v6v  * rv, rstd[13], nms[13]);
      op[14] = fmaf(v12v * rv, rstd[14], nms[14]);
      op[15] = fmaf(f15, rstd[15], nms[15]);
      op[16] = t[8];                       // normalized hour_sin (per-s constant)
      op[17] = t[9];                       // normalized hour_cos
      op[18] = fmaf(mo3, rstd[18], nms[18]);
      op[19] = fmaf(mo6, rstd[19], nms[19]);
      op[20] = fmaf(f20, rstd[20], nms[20]);
      const float r21 = fmaf(rsi, rstd[21], nms[21]);
      op[21] = r21;                        // s=0 slot fixed up below
      if (s == 1) orow[21] = r21;          // edge pad: feature[0] = rsi_raw[0]
      op[22] = fmaf(obv, rstd[22], nms[22]);
      op[23] = 0.0f;                       // broadcast channels: var==0 -> normalized 0
      op[24] = 0.0f;
    }

    lc6 = lc5; lc5 = lc4; lc4 = lc3; lc3 = lc2; lc2 = lc1; lc1 = lc;
    cprev = c;
  }
}

__global__ __launch_bounds__(NT)
void feat_in_kernel(const float* __restrict__ x, float* __restrict__ out, int B) {
  __shared__ __align__(16) float  lds_in[NT * RS];     // TDM-padded 128x120 f32 tile
  __shared__ __align__(16) float  tab[24 * 12];        // per-step wave-uniform constants
  __shared__ __align__(16) float2 hist[24 * NT];       // per-thread gain/loss cumsums

  const int tid  = threadIdx.x;
  const int row0 = blockIdx.x * NT;
  const int row  = row0 + tid;

#if __has_builtin(__builtin_amdgcn_tensor_load_to_lds)
  if ((tid >> 5) == 0) {
    unsigned long long ga = (unsigned long long)(uintptr_t)(x + (size_t)row0 * 120);
    unsigned int la = (unsigned int)(uintptr_t)&lds_in[0];
    v4u g0; v8i g1;
    v4i g2 = {0, 0, 0, 0}; v4i g3 = {0, 0, 0, 0};
    v8i g4 = {0, 0, 0, 0, 0, 0, 0, 0};
    g0.x = 1u;                                     // count=1
    g0.y = la;                                     // lds_addr
    g0.z = (unsigned int)ga;                       // global_addr[31:0]
    g0.w = (unsigned int)(ga >> 32) | (2u << 30);  // global_addr[56:32] | type=2
    const unsigned int td0 = 120, tile0 = 120, tile1 = NT;
    const unsigned long long td1 = (unsigned long long)B;
    const unsigned long long st0 = 120;
    // data_size=4B | pad_enable | pad_interval=8dw(code 2) | pad_amount=1dw(code 0)
    g1[0] = (int)((2u << 16) | (1u << 20) | (2u << 22));
    g1[1] = (int)((td0 & 0xFFFFu) << 16);
    g1[2] = (int)((td0 >> 16) | ((unsigned)(td1 & 0xFFFFu) << 16));
    g1[3] = (int)(((td1 >> 16) & 0xFFFFu) | (tile0 << 16));
    g1[4] = (int)(tile1 & 0xFFFFu);
    g1[5] = (int)(st0 & 0xFFFFFFFFu);
    g1[6] = 0;
    g1[7] = 0;
    __builtin_amdgcn_tensor_load_to_lds(g0, g1, g2, g3, g4, 0);
  }
#else
  for (int k = tid; k < NT * 120; k += NT) {
    const int r = k / 120, e = k % 120;
    const size_t gi = (size_t)row0 * 120 + k;
    lds_in[r * RS + e + (e >> 3)] = (gi < (size_t)B * 120) ? x[gi] : 0.0f;
  }
#endif

  if (tid < 24) {   // per-step constants (q^s, a/(1-q^(s+1)), normalized hour sin/cos)
    const float s = (float)tid;
    const float p3  = exp2f(-s);
    const float p6  = exp2f(-0.48542683f * s);
    const float p12 = exp2f(-0.24100808f * s);
    const float p24 = exp2f(-0.12029423f * s);
    float* t = &tab[tid * 12];
    t[0] = p3;  t[1] = p6;  t[2] = p12;  t[3] = p24;
    t[4] = 0.5f            / (1.0f - p3  * 0.5f);
    t[5] = (2.0f / 7.0f)   / (1.0f - p6  * (5.0f / 7.0f));
    t[6] = (2.0f / 13.0f)  / (1.0f - p12 * (11.0f / 13.0f));
    t[7] = (2.0f / 25.0f)  / (1.0f - p24 * (23.0f / 25.0f));
    const float ang = s * 0.26179938779914946f;     // 2*pi/24
    const float rn  = rsqrtf(0.5f + 1e-5f);         // hour channels: mean=0, var=0.5
    t[8] = sinf(ang) * rn;  t[9] = cosf(ang) * rn;
    t[10] = 0.0f; t[11] = 0.0f;
  }

#if __has_builtin(__builtin_amdgcn_tensor_load_to_lds)
  if ((tid >> 5) == 0) __builtin_amdgcn_s_wait_tensorcnt(0);
#endif
  __syncthreads();

  if (row >= B) return;

  const float* rin = &lds_in[tid * RS];
  float2* histp = &hist[tid];
  float* orow = out + (size_t)row * 600;

  float sum[25], sq[25], nms[25], rstd[25];
#pragma unroll
  for (int ch = 0; ch < 25; ++ch) { sum[ch] = 0.0f; sq[ch] = 0.0f; nms[ch] = 0.0f; rstd[ch] = 0.0f; }

  run_pass<false>(rin, tab, histp, sum, sq, nms, rstd, orow);

#pragma unroll
  for (int ch = 0; ch < 25; ++ch) {
    const float m = sum[ch] * (1.0f / 24.0f);
    const float var = sq[ch] * (1.0f / 24.0f) - m * m;
    const float rs_ = rsqf(var + 1e-5f);
    rstd[ch] = rs_;
    nms[ch] = -m * rs_;          // folded: out = f*rstd - mean*rstd
  }

  run_pass<true>(rin, tab, histp, sum, sq, nms, rstd, orow);
}

extern "C" void kernel_launch(void* const* d_in, const int* in_sizes, int n_in,
                              void* d_out, int out_size, void* d_ws, size_t ws_size,
                              hipStream_t stream) {
  const float* x = (const float*)d_in[0];
  float* out = (float*)d_out;
  const int B = in_sizes[0] / 120;           // (B, 24, 5) f32
  const int grid = (B + NT - 1) / NT;        // 128 rows (1 per thread) per block
  feat_in_kernel<<<grid, NT, 0, stream>>>(x, out, B);
}